// CLVModel_37847251812435
// MI455X (gfx1250) — compile-verified
//
#include <hip/hip_runtime.h>

typedef float v2f __attribute__((ext_vector_type(2)));
typedef float v8f __attribute__((ext_vector_type(8)));

#define KP 148           // padded combined K (x | agg | h)
#define NGATE_COLS 256   // 4 gates x 64
#define MBLK 32          // rows per workgroup

// ---------------------------------------------------------------- utilities
__global__ void zero_kernel(float* p, size_t n) {
    size_t i = (size_t)blockIdx.x * blockDim.x + threadIdx.x;
    size_t stride = (size_t)gridDim.x * blockDim.x;
    for (; i < n; i += stride) p[i] = 0.0f;
}

struct PackArgs {
    const float* Wx[4];  // per gate i,f,g,o : (xin x 64)
    const float* Wl[4];  // (64 x 64)
    const float* Wr[4];  // (64 x 64)
    const float* bl[4];  // (64)
    const float* bg[4];  // (64)
};

// Builds Wcomb[KP][256] (K rows: x | Wl | Wr | zero-pad) and bias[256] = bl+b
__global__ void pack_kernel(PackArgs pa, int xin, float* __restrict__ Wcomb,
                            float* __restrict__ bias) {
    int i = blockIdx.x * blockDim.x + threadIdx.x;
    if (i < NGATE_COLS) {
        int g = i >> 6, n = i & 63;
        bias[i] = pa.bl[g][n] + pa.bg[g][n];
    }
    if (i < KP * NGATE_COLS) {
        int k = i / NGATE_COLS, col = i % NGATE_COLS;
        int g = col >> 6, n = col & 63;
        float v = 0.0f;
        if (k < xin)            v = pa.Wx[g][k * 64 + n];
        else if (k < xin + 64)  v = pa.Wl[g][(k - xin) * 64 + n];
        else if (k < xin + 128) v = pa.Wr[g][(k - xin - 64) * 64 + n];
        Wcomb[i] = v;
    }
}

// ------------------------------------------------------------- aggregation
__global__ void count_kernel(const int* __restrict__ dst, float* __restrict__ cnt, int E) {
    int i = blockIdx.x * blockDim.x + threadIdx.x;
    if (i < E) unsafeAtomicAdd(&cnt[dst[i]], 1.0f);
}

// one thread per (edge, column); 64 consecutive threads share a row -> coalesced
__global__ void scatter_kernel(const int* __restrict__ src, const int* __restrict__ dst,
                               const float* __restrict__ hsrc, float* __restrict__ agg,
                               int total) {
    int i = blockIdx.x * blockDim.x + threadIdx.x;
    if (i >= total) return;
    int e = i >> 6, c = i & 63;
    unsafeAtomicAdd(&agg[(size_t)dst[e] * 64 + c], hsrc[(size_t)src[e] * 64 + c]);
}

// -------------------------------------------------- fused gate GEMM + LSTM
// A = [x_raw | emb | agg/deg | h] (KP cols, zero padded), B = Wcomb[KP][256].
// Block: 256 threads = 8 waves -> 2 m-tiles (16 rows) x 4 col-tiles (16 cols of 64).
// Gate loop stages per-gate B slice (148x64) in LDS; fp32 WMMA 16x16x4, 37 K-steps.
__global__ __launch_bounds__(256)
void fused_gates_kernel(const float* __restrict__ x, int xraw,
                        const float* __restrict__ emb, int embd,
                        const float* __restrict__ agg, const float* __restrict__ cnt,
                        const float* __restrict__ h, const float* __restrict__ cst,
                        const float* __restrict__ Wcomb, const float* __restrict__ bias,
                        float* __restrict__ out_h, float* __restrict__ out_c, int nrows) {
    __shared__ float A_s[MBLK * KP];   // 18,944 B
    __shared__ float B_s[KP * 64];     // 37,888 B

    const int tid = threadIdx.x;
    const int rowbase = blockIdx.x * MBLK;
    const int xin = xraw + embd;
    const int xstride = xraw + 1;

    // assemble A block (zero-padded rows/cols)
    for (int i = tid; i < MBLK * KP; i += 256) {
        int r = i / KP, k = i - r * KP;
        int m = rowbase + r;
        float v = 0.0f;
        if (m < nrows) {
            if (k < xraw) {
                v = x[(size_t)m * xstride + k];
            } else if (k < xin) {
                int id = (int)x[(size_t)m * xstride + xraw];
                v = emb[(size_t)id * embd + (k - xraw)];
            } else if (k < xin + 64) {
                float inv = 1.0f / fmaxf(cnt[m], 1.0f);
                v = agg[(size_t)m * 64 + (k - xin)] * inv;
            } else if (k < xin + 128) {
                v = h[(size_t)m * 64 + (k - xin - 64)];
            }
        }
        A_s[i] = v;
    }

    const int lane = tid & 31;
    const int w = tid >> 5;
    const int mt = w >> 2;              // m-tile 0..1
    const int ct = w & 3;               // col-tile 0..3
    const int ln = lane & 15;
    const int half = lane >> 4;
    const int kh = half * 2;            // K offset for upper lane half
    const int arow = mt * 16 + ln;      // A row this lane supplies
    const int cn = ct * 16 + ln;        // column within 64-wide gate block

    v8f accs[4];
#pragma unroll
    for (int g = 0; g < 4; ++g) {
        __syncthreads();
        // stage gate-g weight slice: B_s[k][n] = Wcomb[k][g*64+n]
        for (int i = tid; i < KP * 64; i += 256) {
            int k = i >> 6, n = i & 63;
            B_s[i] = Wcomb[k * NGATE_COLS + g * 64 + n];
        }
        __syncthreads();

        v8f acc = {0.f, 0.f, 0.f, 0.f, 0.f, 0.f, 0.f, 0.f};
        const float* Arow = &A_s[arow * KP];
#pragma unroll 4
        for (int k0 = 0; k0 < KP; k0 += 4) {
            // A frag: lane holds (m = lane%16, k = k0 + j + 2*(lane/16)), j=0,1
            float2 av = *reinterpret_cast<const float2*>(Arow + k0 + kh);
            v2f a; a.x = av.x; a.y = av.y;
            // B frag: lane holds (k = k0 + j + 2*(lane/16), n = lane%16)
            const float* Bk = &B_s[(k0 + kh) * 64];
            v2f b; b.x = Bk[cn]; b.y = Bk[64 + cn];
            acc = __builtin_amdgcn_wmma_f32_16x16x4_f32(
                false, a, false, b, (short)0, acc, false, false);
        }
        accs[g] = acc;
    }

    // LSTM epilogue directly on accumulator layout: VGPR j -> row j + 8*half
#pragma unroll
    for (int j = 0; j < 8; ++j) {
        int m = rowbase + mt * 16 + j + 8 * half;
        if (m >= nrows) continue;
        float pi = accs[0][j] + bias[cn];
        float pf = accs[1][j] + bias[64 + cn];
        float pg = accs[2][j] + bias[128 + cn];
        float po = accs[3][j] + bias[192 + cn];
        float gi = 1.0f / (1.0f + expf(-pi));
        float gf = 1.0f / (1.0f + expf(-pf));
        float gt = tanhf(pg);
        float go = 1.0f / (1.0f + expf(-po));
        float cold = cst[(size_t)m * 64 + cn];
        float cnew = gf * cold + gi * gt;
        float hnew = go * tanhf(cnew);
        out_c[(size_t)m * 64 + cn] = cnew;
        out_h[(size_t)m * 64 + cn] = hnew;
    }
}

// ----------------------------------------------------------------- pred MLP
__global__ void pred_kernel(const float* __restrict__ hC, const float* __restrict__ W1,
                            const float* __restrict__ b1, const float* __restrict__ W2,
                            const float* __restrict__ b2, float* __restrict__ pred,
                            int nrows) {
    int m = blockIdx.x * blockDim.x + threadIdx.x;
    if (m >= nrows) return;
    float acc[32];
#pragma unroll
    for (int n = 0; n < 32; ++n) acc[n] = b1[n];
    const float* hr = hC + (size_t)m * 64;
    for (int k = 0; k < 64; ++k) {
        float hv = hr[k];
#pragma unroll
        for (int n = 0; n < 32; ++n) acc[n] = fmaf(hv, W1[k * 32 + n], acc[n]);
    }
    float s = b2[0];
#pragma unroll
    for (int n = 0; n < 32; ++n) s += fmaxf(acc[n], 0.0f) * W2[n];
    pred[m] = s;
}

// ---------------------------------------------------------------- launcher
extern "C" void kernel_launch(void* const* d_in, const int* in_sizes, int n_in,
                              void* d_out, int out_size, void* d_ws, size_t ws_size,
                              hipStream_t stream) {
    // input indices (recursive insertion order of setup_inputs):
    // 0 x_customer, 1 x_product, 2 h_customer, 3 h_product, 4 c_customer, 5 c_product
    // 6 country_emb, 7 desc_emb
    // W (i,f,g,o)x(customer,product): 8..15 ; b: 16..23
    // sage (i,f,g,o)x(buys,rev)x(Wl,bl,Wr): 24..47
    // pred W1=48 b1=49 W2=50 b2=51 ; edges 52..55
    const float* x_c   = (const float*)d_in[0];
    const float* x_p   = (const float*)d_in[1];
    const float* h_c   = (const float*)d_in[2];
    const float* h_p   = (const float*)d_in[3];
    const float* c_c   = (const float*)d_in[4];
    const float* c_p   = (const float*)d_in[5];
    const float* cemb  = (const float*)d_in[6];
    const float* demb  = (const float*)d_in[7];
    const int* src_buys = (const int*)d_in[52];
    const int* dst_buys = (const int*)d_in[53];
    const int* src_rev  = (const int*)d_in[54];
    const int* dst_rev  = (const int*)d_in[55];

    const int NC = in_sizes[2] / 64;
    const int NP = in_sizes[3] / 64;
    const int E  = in_sizes[52];

    // workspace layout
    float* ws     = (float*)d_ws;
    float* agg_c  = ws;                              // NC*64
    float* agg_p  = agg_c + (size_t)NC * 64;         // NP*64
    float* cnt_c  = agg_p + (size_t)NP * 64;         // NC
    float* cnt_p  = cnt_c + NC;                      // NP
    float* Wc     = cnt_p + NP;                      // KP*256
    float* Wp     = Wc + KP * NGATE_COLS;            // KP*256
    float* bias_c = Wp + KP * NGATE_COLS;            // 256
    float* bias_p = bias_c + NGATE_COLS;             // 256

    size_t zn = (size_t)NC * 64 + (size_t)NP * 64 + NC + NP;
    zero_kernel<<<2048, 256, 0, stream>>>(agg_c, zn);

    PackArgs pc, pp;
    for (int g = 0; g < 4; ++g) {
        int sbase = 24 + 6 * g;
        pc.Wx[g] = (const float*)d_in[8 + 2 * g];    // W.g.customer
        pc.Wl[g] = (const float*)d_in[sbase + 3];    // sage.g.rev.Wl
        pc.bl[g] = (const float*)d_in[sbase + 4];    // sage.g.rev.bl
        pc.Wr[g] = (const float*)d_in[sbase + 5];    // sage.g.rev.Wr
        pc.bg[g] = (const float*)d_in[16 + 2 * g];   // b.g.customer
        pp.Wx[g] = (const float*)d_in[9 + 2 * g];    // W.g.product
        pp.Wl[g] = (const float*)d_in[sbase + 0];    // sage.g.buys.Wl
        pp.bl[g] = (const float*)d_in[sbase + 1];    // sage.g.buys.bl
        pp.Wr[g] = (const float*)d_in[sbase + 2];    // sage.g.buys.Wr
        pp.bg[g] = (const float*)d_in[17 + 2 * g];   // b.g.product
    }
    int packBlocks = (KP * NGATE_COLS + 255) / 256;
    pack_kernel<<<packBlocks, 256, 0, stream>>>(pc, 18, Wc, bias_c);
    pack_kernel<<<packBlocks, 256, 0, stream>>>(pp, 20, Wp, bias_p);

    count_kernel<<<(E + 255) / 256, 256, 0, stream>>>(dst_rev, cnt_c, E);
    count_kernel<<<(E + 255) / 256, 256, 0, stream>>>(dst_buys, cnt_p, E);
    int etot = E * 64;
    scatter_kernel<<<(etot + 255) / 256, 256, 0, stream>>>(src_rev, dst_rev, h_p, agg_c, etot);
    scatter_kernel<<<(etot + 255) / 256, 256, 0, stream>>>(src_buys, dst_buys, h_c, agg_p, etot);

    // output layout: pred | h_c_new | h_p_new | c_c_new | c_p_new
    float* out      = (float*)d_out;
    float* pred     = out;
    float* h_c_new  = pred + NC;
    float* h_p_new  = h_c_new + (size_t)NC * 64;
    float* c_c_new  = h_p_new + (size_t)NP * 64;
    float* c_p_new  = c_c_new + (size_t)NC * 64;

    fused_gates_kernel<<<(NC + MBLK - 1) / MBLK, 256, 0, stream>>>(
        x_c, 14, cemb, 4, agg_c, cnt_c, h_c, c_c, Wc, bias_c, h_c_new, c_c_new, NC);
    fused_gates_kernel<<<(NP + MBLK - 1) / MBLK, 256, 0, stream>>>(
        x_p, 4, demb, 16, agg_p, cnt_p, h_p, c_p, Wp, bias_p, h_p_new, c_p_new, NP);

    pred_kernel<<<(NC + 255) / 256, 256, 0, stream>>>(
        h_c_new, (const float*)d_in[48], (const float*)d_in[49],
        (const float*)d_in[50], (const float*)d_in[51], pred, NC);
}